// OctreeImportanceRenderer_76948634075583
// MI455X (gfx1250) — compile-verified
//
#include <hip/hip_runtime.h>

// ---- constants mirroring the reference ----
#define N_FINE_PTS   2000000
#define N_COARSE_PTS 150000
#define RES_F  256
#define RES_C  64
#define FEAT   32
#define NCH    (FEAT + 1)     // 33 floats per feature row
#define S_COARSE 48
#define FINE_RES 16
#define S_ALL    64
#define R_RAYS   16384
#define C_OUT    65           // 2*FEAT + 1
#define RAYS_PER_BLOCK 8
#define NBUF 8                // LDS staging ring slots
#define PIPE 4                // async pipeline depth (2 async ops per stage)

// ---- CDNA5 async global->LDS (ASYNCcnt) with builtin/asm fallback ----
#if __has_builtin(__builtin_amdgcn_s_wait_asynccnt)
#define WAIT_ASYNC(n) __builtin_amdgcn_s_wait_asynccnt((unsigned short)(n))
#else
#define WAIT_ASYNC(n) asm volatile("s_wait_asynccnt %0" :: "i"(n) : "memory")
#endif

__device__ __forceinline__ void async_copy_f32(const float* gsrc, float* lds_dst) {
#if __has_builtin(__builtin_amdgcn_global_load_async_to_lds_b32)
    // prototype: (global int* src, local int* dst, imm offset, imm cpol)
    __builtin_amdgcn_global_load_async_to_lds_b32(
        (__attribute__((address_space(1))) int*)(uintptr_t)gsrc,
        (__attribute__((address_space(3))) int*)(uint32_t)(uintptr_t)lds_dst,
        0, 0);
#else
    // per-lane: copies one dword from global gsrc into LDS at lds_dst; tracked by ASYNCcnt
    asm volatile("global_load_async_to_lds_b32 %0, %1, off"
                 :: "v"((uint32_t)(uintptr_t)lds_dst),
                    "v"((unsigned long long)(uintptr_t)gsrc)
                 : "memory");
#endif
}

__device__ __forceinline__ float softplusf(float x) {
    // jax.nn.softplus, numerically stable
    return fmaxf(x, 0.0f) + log1pf(expf(-fabsf(x)));
}

struct RayScratch {
    float depth[S_ALL];
    float dens[S_ALL];
    int   rowf[S_ALL];
    int   rowc[S_ALL];
    float w[S_ALL - 1];
    float alp[S_ALL - 1];
    float dmd[S_ALL - 1];
    float stageF[NBUF][32];   // async-staged fine rows   (channel = lane)
    float stageC[NBUF][32];   // async-staged coarse rows
};

// One octree level: voxel hash -> grid -> feature-row index + density.
__device__ __forceinline__ void query_level(
    float s0, float s1, float s2, int res,
    const int* __restrict__ grid, const float* __restrict__ feats,
    int offset, int npts, float& dens, int& row, bool& valid_out)
{
    const float fres = (float)res;
    int v0 = (int)floorf((s0 + 1.0f) * 0.5f * fres);
    int v1 = (int)floorf((s1 + 1.0f) * 0.5f * fres);
    int v2 = (int)floorf((s2 + 1.0f) * 0.5f * fres);
    bool inb = (v0 >= 0) && (v0 < res) && (v1 >= 0) && (v1 < res) &&
               (v2 >= 0) && (v2 < res);
    int c0 = min(max(v0, 0), res - 1);
    int c1 = min(max(v1, 0), res - 1);
    int c2 = min(max(v2, 0), res - 1);
    int flat = (c0 * res + c1) * res + c2;
    int pi = grid[flat] - offset;
    bool valid = inb && (pi >= 0);
    int pu = valid ? pi : 0;
    pu = min(pu, npts - 1);                 // jax OOB-gather clamps
    bool m = valid && (pu != 0);            // feat_collected[point_index==0] = 0
    dens = m ? feats[(size_t)pu * NCH + FEAT] : 0.0f;
    row  = m ? pu : -1;                     // -1 encodes "contributes zero"
    valid_out = valid;
}

__device__ __forceinline__ void query_point(
    float px, float py, float pz,
    const int* __restrict__ fg, const float* __restrict__ ff,
    const int* __restrict__ cg, const float* __restrict__ cf,
    float& density, int& rowf, int& rowc)
{
    const float s0 = px * (1.0f / 50.0f);          // XY_BOUND
    const float s1 = py * (1.0f / 50.0f);
    const float s2 = (pz - 5.0f) * (1.0f / 10.0f); // Z_CENTER / Z_HALF
    float fd, cd; bool fv, cv;
    query_level(s0, s1, s2, RES_F, fg, ff, N_FINE_PTS,   N_FINE_PTS,   fd, rowf, fv);
    query_level(s0, s1, s2, RES_C, cg, cf, N_COARSE_PTS, N_COARSE_PTS, cd, rowc, cv);
    // density = fvalid ? fdens : (cvalid ? cdens : -DENSITY_CLAMP)
    float d = fv ? fd : (cv ? cd : -10.0f);
    // soft_clamp(x, RENDER_CLAMP=10)
    density = tanhf(d * 0.1f) * 10.0f;
}

__global__ __launch_bounds__(256)
void OctreeImportanceRenderer_76948634075583_kernel(
    const float* __restrict__ origins,
    const float* __restrict__ dirs,
    const int*   __restrict__ fine_grid,
    const float* __restrict__ fine_feats,
    const int*   __restrict__ coarse_grid,
    const float* __restrict__ coarse_feats,
    float* __restrict__ out)
{
    __shared__ RayScratch sh_all[RAYS_PER_BLOCK];
    const int wave = threadIdx.x >> 5;       // wave32: one wave per ray
    const int lane = threadIdx.x & 31;
    const int r = blockIdx.x * RAYS_PER_BLOCK + wave;
    RayScratch* sh = &sh_all[wave];

    const float ox = origins[3 * r + 0], oy = origins[3 * r + 1], oz = origins[3 * r + 2];
    const float dx = dirs[3 * r + 0],    dy = dirs[3 * r + 1],    dz = dirs[3 * r + 2];
    const float CSTEP = (60.0f - 0.5f) / 47.0f;   // linspace(NEAR, MAX_DEPTH, 48) step

    // ---- Phase A: coarse densities (colors never affect coarse weights) ----
    for (int s = lane; s < S_COARSE; s += 32) {
        float t = 0.5f + (float)s * CSTEP;
        float density; int rf, rc;
        query_point(ox + t * dx, oy + t * dy, oz + t * dz,
                    fine_grid, fine_feats, coarse_grid, coarse_feats,
                    density, rf, rc);
        sh->dens[s] = density;
    }
    __syncthreads();

    // ---- Phase A scan + Phase B merge (lane 0, cheap serial scan) ----
    if (lane == 0) {
        float T = 1.0f, best = -1.0f; int bi = 0;
        for (int i = 0; i < S_COARSE - 1; ++i) {
            float dm = 0.5f * (sh->dens[i] + sh->dens[i + 1]);
            float sg = softplusf(dm - 1.0f);
            float a  = 1.0f - expf(-sg * CSTEP);
            float wg = a * T;
            if (wg > best) { best = wg; bi = i; }   // first-max = jnp.argmax
            T *= (1.0f - a + 1e-10f);
        }
        float mwd = 0.5f + (float)bi * CSTEP;       // take_along_axis(depths, argmax)
        // Merge 48 sorted coarse + 16 sorted (clipped) fine depths.
        // Coarse-first on ties == stable argsort of concat([coarse, fine]).
        int a = 0, b = 0;
        for (int t = 0; t < S_ALL; ++t) {
            float ca = (a < S_COARSE) ? (0.5f + (float)a * CSTEP) : 3.0e38f;
            float fb = (b < FINE_RES)
                     ? fminf(fmaxf(mwd + (float)(b - 8) * 0.2f, 0.0f), 60.0f)
                     : 3.0e38f;
            if (ca <= fb) { sh->depth[t] = ca; ++a; }
            else          { sh->depth[t] = fb; ++b; }
        }
    }
    __syncthreads();

    // ---- Phase C: densities + feature-row ids for the 64 sorted samples ----
    for (int j = lane; j < S_ALL; j += 32) {
        float t = sh->depth[j];
        float density; int rf, rc;
        query_point(ox + t * dx, oy + t * dy, oz + t * dz,
                    fine_grid, fine_feats, coarse_grid, coarse_feats,
                    density, rf, rc);
        sh->dens[j] = density;
        sh->rowf[j] = rf;
        sh->rowc[j] = rc;
    }
    __syncthreads();

    // Issue helper: stage feature rows for sample p into the LDS ring via the
    // CDNA5 async global->LDS engine. Invalid rows fetch row 0 and are zero-
    // masked at consume time (coefficient = 0), keeping issue unconditional.
    auto issue = [&](int p) {
        int rf = sh->rowf[p]; int rfi = rf >= 0 ? rf : 0;
        int rc = sh->rowc[p]; int rci = rc >= 0 ? rc : 0;
        int slot = p & (NBUF - 1);
        async_copy_f32(&fine_feats[(size_t)rfi * NCH + lane],   &sh->stageF[slot][lane]);
        async_copy_f32(&coarse_feats[(size_t)rci * NCH + lane], &sh->stageC[slot][lane]);
    };

    // Prologue: start the gather pipeline NOW so HBM latency overlaps the
    // serial transmittance scan below.
    for (int p = 0; p < PIPE; ++p) issue(p);

    // ---- Final ray-march scan (lane 0): transmittance cumprod ----
    if (lane == 0) {
        float T = 1.0f, df = 0.0f;
        for (int i = 0; i < S_ALL - 1; ++i) {
            float d0 = sh->depth[i], d1 = sh->depth[i + 1];
            float delta = d1 - d0;
            float dmd   = 0.5f * (d0 + d1);
            float dm    = 0.5f * (sh->dens[i] + sh->dens[i + 1]);
            float sg = softplusf(dm - 1.0f);
            float a  = 1.0f - expf(-sg * delta);
            float wg = a * T;
            df = fmaf(wg, dmd, df);
            sh->w[i] = wg; sh->alp[i] = a; sh->dmd[i] = dmd;
            T *= (1.0f - a + 1e-10f);
        }
        out[(size_t)R_RAYS * C_OUT + (size_t)r] = df;   // depth_final
        out[(size_t)r * C_OUT + 64] = 0.0f;             // rgb channel 64 == 0
    }
    __syncthreads();

    // ---- Emit weights / alpha / depths_mid (coalesced across lanes) ----
    {
        const size_t OFF_W = (size_t)R_RAYS * C_OUT + (size_t)R_RAYS;
        const size_t OFF_A = OFF_W + (size_t)R_RAYS * 63;
        const size_t OFF_D = OFF_A + (size_t)R_RAYS * 63;
        for (int i = lane; i < S_ALL - 1; i += 32) {
            out[OFF_W + (size_t)r * 63 + i] = sh->w[i];
            out[OFF_A + (size_t)r * 63 + i] = sh->alp[i];
            out[OFF_D + (size_t)r * 63 + i] = sh->dmd[i];
        }
    }

    // ---- Feature accumulation through the async LDS ring.
    //      Async loads complete in order; with 2 ops/stage and PIPE stages in
    //      flight (8 outstanding at steady state), s_wait_asynccnt <= 6
    //      retires exactly the oldest stage. rgb[c] = sum_j g[j]*col[j,c],
    //      g[j] = 0.5*(w[j-1]+w[j]).
    float accf = 0.0f, accc = 0.0f;
    auto consume = [&](int j) {
        float wl = (j > 0)         ? sh->w[j - 1] : 0.0f;
        float wr = (j < S_ALL - 1) ? sh->w[j]     : 0.0f;
        float gj = 0.5f * (wl + wr);
        int slot = j & (NBUF - 1);
        float fvv = sh->stageF[slot][lane];
        float cvv = sh->stageC[slot][lane];
        accf = fmaf((sh->rowf[j] >= 0) ? gj : 0.0f, fvv, accf);
        accc = fmaf((sh->rowc[j] >= 0) ? gj : 0.0f, cvv, accc);
    };
    // Steady state: j = 0..S_ALL-PIPE (8 ops outstanding at entry).
    for (int j = 0; j <= S_ALL - PIPE; ++j) {
        WAIT_ASYNC(2 * (PIPE - 1));
        consume(j);
        if (j + PIPE < S_ALL) issue(j + PIPE);
    }
    // Drain: remaining <= 6 ops belong to the last PIPE-1 stages; wait for all
    // of them once, then consume the tail without further waits.
    WAIT_ASYNC(0);
    for (int j = S_ALL - PIPE + 1; j < S_ALL; ++j) consume(j);

    out[(size_t)r * C_OUT + lane]      = accf;  // channels 0..31  (fine feats)
    out[(size_t)r * C_OUT + 32 + lane] = accc;  // channels 32..63 (coarse feats)
}

extern "C" void kernel_launch(void* const* d_in, const int* in_sizes, int n_in,
                              void* d_out, int out_size, void* d_ws, size_t ws_size,
                              hipStream_t stream) {
    const float* origins      = (const float*)d_in[0];
    const float* dirs         = (const float*)d_in[1];
    const int*   fine_grid    = (const int*)d_in[2];
    const float* fine_feats   = (const float*)d_in[3];
    const int*   coarse_grid  = (const int*)d_in[4];
    const float* coarse_feats = (const float*)d_in[5];
    float* out = (float*)d_out;

    dim3 grid(R_RAYS / RAYS_PER_BLOCK);   // 2048 blocks
    dim3 block(256);                      // 8 wave32 = 8 rays per block
    hipLaunchKernelGGL(OctreeImportanceRenderer_76948634075583_kernel,
                       grid, block, 0, stream,
                       origins, dirs, fine_grid, fine_feats,
                       coarse_grid, coarse_feats, out);
}